// SpatioConvLayer_70420283785458
// MI455X (gfx1250) — compile-verified
//
#include <hip/hip_runtime.h>
#include <math.h>

// ---- problem constants (from reference setup_inputs) ----
#define B_  2
#define N_  10000
#define T_  8
#define H_  2
#define C_  32
#define O_  (H_ * C_)        // 64
#define R_  (N_ * B_ * T_)   // 160000 rows of the dense GEMM
#define TH_ (T_ * H_)        // 16
#define NEG_SLOPE 0.2f

typedef __attribute__((ext_vector_type(2))) float v2f;
typedef __attribute__((ext_vector_type(8))) float v8f;

// ---------------- atomic helpers (native f32 L2 atomics) ----------------
__device__ __forceinline__ void atomicAddF(float* addr, float val) {
  __hip_atomic_fetch_add(addr, val, __ATOMIC_RELAXED, __HIP_MEMORY_SCOPE_AGENT);
}

// float max via monotone int/uint ordering trick (works with -inf init)
__device__ __forceinline__ void atomicMaxF(float* addr, float val) {
  if (val >= 0.0f) {
    __hip_atomic_fetch_max((int*)addr, __float_as_int(val),
                           __ATOMIC_RELAXED, __HIP_MEMORY_SCOPE_AGENT);
  } else {
    __hip_atomic_fetch_min((unsigned int*)addr, __float_as_uint(val),
                           __ATOMIC_RELAXED, __HIP_MEMORY_SCOPE_AGENT);
  }
}

// ---------------- kernels ----------------
__global__ void k_zero(float* __restrict__ p, int n) {
  int i = blockIdx.x * blockDim.x + threadIdx.x;
  if (i < n) p[i] = 0.0f;
}

__global__ void k_init_stats(float* __restrict__ m, float* __restrict__ denom, int n) {
  int i = blockIdx.x * blockDim.x + threadIdx.x;
  if (i < n) { m[i] = -__builtin_inff(); denom[i] = 0.0f; }
}

// feat[(n*B+b)*T+t, o] = sum_c x[(b*N+n)*T+t, c] * W[o, c]
// One wave per 16-row tile; fp32 WMMA 16x16x4, K=32 -> 8 steps, 4 col tiles.
__global__ void __launch_bounds__(256)
k_gemm_feat_wmma(const float* __restrict__ x, const float* __restrict__ W,
                 float* __restrict__ feat) {
  const int wave = blockIdx.x * (blockDim.x >> 5) + (threadIdx.x >> 5);
  const int lane = threadIdx.x & 31;
  const int row0 = wave * 16;
  if (row0 >= R_) return;

  const int mcol  = lane & 15;        // M for A-frag / N for B,D-frags
  const int khalf = (lane >> 4) * 2;  // 0 or 2: K sub-offset within a K=4 step

  const float* __restrict__ xrow = x + (size_t)(row0 + mcol) * C_;

  v8f acc0 = {}, acc1 = {}, acc2 = {}, acc3 = {};
#pragma unroll
  for (int s = 0; s < 8; ++s) {
    const int k = 4 * s + khalf;
    v2f a; a.x = xrow[k]; a.y = xrow[k + 1];
    // B-frag: lane holds W[o = 16j + mcol][k], W[o][k+1]
    v2f b0; b0.x = W[(0  + mcol) * C_ + k]; b0.y = W[(0  + mcol) * C_ + k + 1];
    v2f b1; b1.x = W[(16 + mcol) * C_ + k]; b1.y = W[(16 + mcol) * C_ + k + 1];
    v2f b2; b2.x = W[(32 + mcol) * C_ + k]; b2.y = W[(32 + mcol) * C_ + k + 1];
    v2f b3; b3.x = W[(48 + mcol) * C_ + k]; b3.y = W[(48 + mcol) * C_ + k + 1];
    acc0 = __builtin_amdgcn_wmma_f32_16x16x4_f32(false, a, false, b0, (short)0, acc0, false, false);
    acc1 = __builtin_amdgcn_wmma_f32_16x16x4_f32(false, a, false, b1, (short)0, acc1, false, false);
    acc2 = __builtin_amdgcn_wmma_f32_16x16x4_f32(false, a, false, b2, (short)0, acc2, false, false);
    acc3 = __builtin_amdgcn_wmma_f32_16x16x4_f32(false, a, false, b3, (short)0, acc3, false, false);
  }

  // D-frag: vgpr i, lane L -> D[M = i + 8*(L/16)][N = L%16]
#pragma unroll
  for (int i = 0; i < 8; ++i) {
    const int mrow = i + 8 * (lane >> 4);
    const int rx   = row0 + mrow;            // x-row in [b,n,t] order
    const int t    = rx % T_;
    const int nb   = rx / T_;
    const int n    = nb % N_;
    const int b    = nb / N_;
    const size_t frow = ((size_t)(n * B_ + b) * T_ + t);
    float* __restrict__ fp = feat + frow * O_ + mcol;
    fp[0]  = acc0[i];
    fp[16] = acc1[i];
    fp[32] = acc2[i];
    fp[48] = acc3[i];
  }
}

// el[r,h] = dot(feat[r, h*C : h*C+C], attn_l[h]);  er likewise.
__global__ void k_el_er(const float* __restrict__ feat,
                        const float* __restrict__ attn_l,
                        const float* __restrict__ attn_r,
                        float* __restrict__ el, float* __restrict__ er) {
  int i = blockIdx.x * blockDim.x + threadIdx.x;   // over R_*H_
  if (i >= R_ * H_) return;
  const int h = i & (H_ - 1);
  const int r = i >> 1;
  const float* __restrict__ f  = feat + (size_t)r * O_ + h * C_;
  const float* __restrict__ al = attn_l + h * C_;
  const float* __restrict__ ar = attn_r + h * C_;
  float sl = 0.0f, sr = 0.0f;
#pragma unroll
  for (int c = 0; c < C_; ++c) { float v = f[c]; sl += v * al[c]; sr += v * ar[c]; }
  el[i] = sl; er[i] = sr;
}

// e = leaky_relu(el[src] + er[dst]); store; atomic segment-max by dst.
__global__ void k_logits_max(const int* __restrict__ src, const int* __restrict__ dst,
                             const float* __restrict__ el, const float* __restrict__ er,
                             float* __restrict__ e_buf, float* __restrict__ m, int total) {
  int i = blockIdx.x * blockDim.x + threadIdx.x;   // over E*B*T*H
  if (i >= total) return;
  const int h = i & 1;
  const int t = (i >> 1) & 7;
  const int b = (i >> 4) & 1;
  const int e = i >> 5;
  const int s = src[e], d = dst[e];
  const int rs = (s * B_ + b) * T_ + t;
  const int rd = (d * B_ + b) * T_ + t;
  float v = el[rs * H_ + h] + er[rd * H_ + h];
  v = v >= 0.0f ? v : NEG_SLOPE * v;
  e_buf[i] = v;
  atomicMaxF(&m[rd * H_ + h], v);
}

// a = exp(e - m[dst]); store; atomic segment-sum of denom by dst.
__global__ void k_exp_denom(const int* __restrict__ dst,
                            float* __restrict__ e_buf, const float* __restrict__ m,
                            float* __restrict__ denom, int total) {
  int i = blockIdx.x * blockDim.x + threadIdx.x;   // over E*B*T*H
  if (i >= total) return;
  const int h = i & 1;
  const int t = (i >> 1) & 7;
  const int b = (i >> 4) & 1;
  const int e = i >> 5;
  const int d = dst[e];
  const int rd = (d * B_ + b) * T_ + t;
  const float a = expf(e_buf[i] - m[rd * H_ + h]);
  e_buf[i] = a;
  atomicAddF(&denom[rd * H_ + h], a);
}

// msg scatter: 32 consecutive lanes share (e,b,t,h), cover c=0..31
// -> coalesced feat reads and coalesced f32 atomic adds into L2.
__global__ void __launch_bounds__(256)
k_scatter(const int* __restrict__ src, const int* __restrict__ dst,
          const float* __restrict__ feat, const float* __restrict__ e_buf,
          const float* __restrict__ denom, float* __restrict__ out, int total) {
  int i = blockIdx.x * blockDim.x + threadIdx.x;   // over E*B*T*H*C
  if (i >= total) return;
  const int c    = i & 31;
  const int rest = i >> 5;        // (e,b,t,h) flat index, same as e_buf index
  const int h = rest & 1;
  const int t = (rest >> 1) & 7;
  const int b = (rest >> 4) & 1;
  const int e = rest >> 5;
  const int s = src[e], d = dst[e];
  const int rs = (s * B_ + b) * T_ + t;
  const int rd = (d * B_ + b) * T_ + t;
  const float alpha = e_buf[rest] / fmaxf(denom[rd * H_ + h], 1e-16f);
  const float msg = feat[(size_t)rs * O_ + h * C_ + c] * alpha;
  const size_t oi = (((size_t)b * N_ + d) * TH_ + t * H_ + h) * C_ + c;
  atomicAddF(&out[oi], msg);
}

// out = relu(out + bias[h*C + c]) in-place, layout [b, n, t*H + h, c]
__global__ void k_bias_relu(float* __restrict__ out, const float* __restrict__ bias, int n) {
  int i = blockIdx.x * blockDim.x + threadIdx.x;
  if (i >= n) return;
  const int c = i & (C_ - 1);
  const int h = (i >> 5) & (H_ - 1);
  float v = out[i] + bias[h * C_ + c];
  out[i] = v > 0.0f ? v : 0.0f;
}

// ---------------- launch ----------------
extern "C" void kernel_launch(void* const* d_in, const int* in_sizes, int n_in,
                              void* d_out, int out_size, void* d_ws, size_t ws_size,
                              hipStream_t stream) {
  const float* x      = (const float*)d_in[0];
  const float* W      = (const float*)d_in[1];
  const float* attn_l = (const float*)d_in[2];
  const float* attn_r = (const float*)d_in[3];
  const float* bias   = (const float*)d_in[4];
  const int*   src    = (const int*)d_in[5];
  const int*   dst    = (const int*)d_in[6];
  const int E = in_sizes[5];
  float* out = (float*)d_out;

  // workspace layout (floats)
  float* feat  = (float*)d_ws;                 // R_*O_      = 10,240,000
  float* el    = feat  + (size_t)R_ * O_;      // R_*H_      =    320,000
  float* er    = el    + (size_t)R_ * H_;
  float* m     = er    + (size_t)R_ * H_;
  float* denom = m     + (size_t)R_ * H_;
  float* e_buf = denom + (size_t)R_ * H_;      // E*B_*T_*H_ = E*32

  const int nbth   = E * B_ * T_ * H_;         // 5,120,000
  const int nbthc  = nbth * C_;                // 163,840,000

  // 1) zero the output accumulator (harness poisons it)
  k_zero<<<(out_size + 255) / 256, 256, 0, stream>>>(out, out_size);
  // 2) init segment max (-inf) and denom (0)
  k_init_stats<<<(R_ * H_ + 255) / 256, 256, 0, stream>>>(m, denom, R_ * H_);
  // 3) dense projection via fp32 WMMA: one wave per 16-row tile
  {
    const int waves = R_ / 16;                 // 10,000
    const int blocks = (waves + 7) / 8;        // 8 waves / 256-thread block
    k_gemm_feat_wmma<<<blocks, 256, 0, stream>>>(x, W, feat);
  }
  // 4) attention dot products
  k_el_er<<<(R_ * H_ + 255) / 256, 256, 0, stream>>>(feat, attn_l, attn_r, el, er);
  // 5) edge logits + segment max
  k_logits_max<<<(nbth + 255) / 256, 256, 0, stream>>>(src, dst, el, er, e_buf, m, nbth);
  // 6) exp + segment sum
  k_exp_denom<<<(nbth + 255) / 256, 256, 0, stream>>>(dst, e_buf, m, denom, nbth);
  // 7) weighted message scatter (coalesced f32 atomics, all tables L2-resident)
  k_scatter<<<(nbthc + 255) / 256, 256, 0, stream>>>(src, dst, feat, e_buf, denom, out, nbthc);
  // 8) bias + relu in place
  k_bias_relu<<<(out_size + 255) / 256, 256, 0, stream>>>(out, bias, out_size);
}